// OIM_MLP_19310172963555
// MI455X (gfx1250) — compile-verified
//
#include <hip/hip_runtime.h>
#include <hip/hip_bf16.h>

// OIM MLP equilibrium dynamics on MI455X (gfx1250).
// fp32 WMMA (v_wmma_f32_16x16x4_f32) for the 256x2048x2048 GEMMs,
// 2-deep software pipeline for fragment loads + deterministic split-K=4;
// VALU for the tiny W2 (K=10 / N=10) couplings.

#define BATCH 256
#define HID   2048
#define OUTD  10
#define EPSS  0.1f
#define BETA  0.5f
#define SLICES 4   // split-K factor (deterministic: separate partial buffers)

typedef __attribute__((ext_vector_type(2))) float v2f;
typedef __attribute__((ext_vector_type(8))) float v8f;

// ---------------------------------------------------------------------------
// WMMA GEMM, split-K over blockIdx.z:
//   C1[z][b,n] = sum_{k in slice z} A1[b,k] * B(k,n)    (+ A2 -> C2 if DUAL)
//   TRANSB=0: B(k,n) = B[k*N + n]   (row-major K x N)
//   TRANSB=1: B(k,n) = B[n*K + k]   (row-major N x K, i.e. W[n, k])
// Wave tile: 16(M) x 64(N). Block: 8 waves -> 16 x 512.
// Grid: (N/512, M/16, SLICES)  -> 2048 waves for N=2048, M=256.
// Two-stage fragment pipeline: loads for k+8 issued before WMMAs for k.
// ---------------------------------------------------------------------------
template <bool TRANSB, bool DUAL>
__global__ __launch_bounds__(256) void gemm_wmma_f32(
    const float* __restrict__ A1, const float* __restrict__ A2,
    const float* __restrict__ B,
    float* __restrict__ C1, float* __restrict__ C2,
    int M, int N, int K)
{
    const int lane = threadIdx.x & 31;
    const int wv   = threadIdx.x >> 5;          // 0..7
    const int m0   = blockIdx.y * 16;
    const int n0   = (blockIdx.x * 8 + wv) * 64;
    if (n0 >= N || m0 >= M) return;

    const int r = lane & 15;                    // row / col within tile
    const int h = lane >> 4;                    // half select (K pairs)

    const int KL = K / SLICES;                  // slice length (512)
    const int K0 = blockIdx.z * KL;
    const size_t cOff = (size_t)blockIdx.z * M * N;

    v8f acc1[4] = {v8f{}, v8f{}, v8f{}, v8f{}};
    v8f acc2[4] = {v8f{}, v8f{}, v8f{}, v8f{}};

    const float* a1p = A1 + (size_t)(m0 + r) * K + K0 + 2 * h;
    const float* a2p = DUAL ? (A2 + (size_t)(m0 + r) * K + K0 + 2 * h) : a1p;

    // fragment loaders (kk is k offset within this slice)
    auto loadA = [&](v2f& d1, v2f& d2, int kk) {
        d1 = *(const v2f*)(a1p + kk);
        d2 = d1;
        if (DUAL) d2 = *(const v2f*)(a2p + kk);
    };
    auto loadB = [&](v2f (&dst)[4], int kk) {
#pragma unroll
        for (int t = 0; t < 4; ++t) {
            const int n = n0 + t * 16 + r;
            if (TRANSB) {
                dst[t] = *(const v2f*)(B + (size_t)n * K + K0 + kk + 2 * h);
            } else {
                dst[t].x = B[(size_t)(K0 + kk + 2 * h) * N + n];
                dst[t].y = B[(size_t)(K0 + kk + 1 + 2 * h) * N + n];
            }
        }
    };
    auto compute = [&](const v2f& a1, const v2f& a2, const v2f (&b)[4]) {
#pragma unroll
        for (int t = 0; t < 4; ++t) {
            acc1[t] = __builtin_amdgcn_wmma_f32_16x16x4_f32(
                false, a1, false, b[t], (short)0, acc1[t], false, false);
            if (DUAL)
                acc2[t] = __builtin_amdgcn_wmma_f32_16x16x4_f32(
                    false, a2, false, b[t], (short)0, acc2[t], false, false);
        }
    };

    // ---- pipeline prologue: stages for k-steps 0 and 4 ----
    v2f a1c, a2c, bc[4];   // fragments for kk
    v2f a1n, a2n, bn[4];   // fragments for kk+4
    loadA(a1c, a2c, 0);
    loadB(bc, 0);
    loadA(a1n, a2n, 4);
    loadB(bn, 4);

    // ---- steady state: issue loads for kk+8, compute kk, rotate ----
    int kk = 0;
    for (; kk < KL - 8; kk += 4) {
        v2f a1m, a2m, bm[4];
        loadA(a1m, a2m, kk + 8);
        loadB(bm, kk + 8);

        compute(a1c, a2c, bc);

        a1c = a1n; a2c = a2n;
#pragma unroll
        for (int t = 0; t < 4; ++t) bc[t] = bn[t];
        a1n = a1m; a2n = a2m;
#pragma unroll
        for (int t = 0; t < 4; ++t) bn[t] = bm[t];
    }

    // ---- epilogue: last two k-steps ----
    compute(a1c, a2c, bc);
    compute(a1n, a2n, bn);

    // C/D layout: VGPR e -> row m0 + e + 8*h, col = n0 + t*16 + r
#pragma unroll
    for (int t = 0; t < 4; ++t) {
        const int col = n0 + t * 16 + r;
#pragma unroll
        for (int e = 0; e < 8; ++e) {
            const int row = m0 + e + 8 * h;
            C1[cOff + (size_t)row * N + col] = acc1[t][e];
            if (DUAL) C2[cOff + (size_t)row * N + col] = acc2[t][e];
        }
    }
}

// ---------------------------------------------------------------------------
// Sum SLICES partial buffers (fixed order -> deterministic)
// ---------------------------------------------------------------------------
__device__ __forceinline__ float sum_slices(const float* __restrict__ p, int i, int n)
{
    float v0 = p[i]               + p[i + (size_t)n];
    float v1 = p[i + (size_t)2*n] + p[i + (size_t)3*n];
    return v0 + v1;
}

__global__ void reduce_slices_kernel(const float* __restrict__ part,
                                     float* __restrict__ out, int n)
{
    int i = blockIdx.x * blockDim.x + threadIdx.x;
    if (i >= n) return;
    out[i] = sum_slices(part, i, n);
}

// ---------------------------------------------------------------------------
// Elementwise trig snapshots
// ---------------------------------------------------------------------------
__global__ void trig_kernel(const float* __restrict__ phi,
                            float* __restrict__ c, float* __restrict__ s, int n)
{
    int i = blockIdx.x * blockDim.x + threadIdx.x;
    if (i >= n) return;
    float p = phi[i];
    c[i] = cosf(p);
    s[i] = sinf(p);
}

// layer-2 trig, padded to stride 16 (cols 10..15 zero)
__global__ void trig2_kernel(const float* __restrict__ phi2,
                             float* __restrict__ c2, float* __restrict__ s2)
{
    int i = blockIdx.x * blockDim.x + threadIdx.x;   // BATCH*16
    if (i >= BATCH * 16) return;
    int b = i >> 4, j = i & 15;
    if (j < OUTD) {
        float p = phi2[b * OUTD + j];
        c2[i] = cosf(p);
        s2[i] = sinf(p);
    } else {
        c2[i] = 0.f;
        s2[i] = 0.f;
    }
}

// ---------------------------------------------------------------------------
// Phase updates (read snapshots + GEMM partials, write phi in place)
// ---------------------------------------------------------------------------
__global__ void update0_kernel(float* __restrict__ phi0,
                               const float* __restrict__ c0, const float* __restrict__ s0,
                               const float* __restrict__ a0,
                               const float* __restrict__ P1cP, const float* __restrict__ P1sP,
                               const float* __restrict__ b0, const float* __restrict__ k0)
{
    int i = blockIdx.x * blockDim.x + threadIdx.x;
    if (i >= BATCH * HID) return;
    int col = i & (HID - 1);
    float p1c = sum_slices(P1cP, i, BATCH * HID);
    float p1s = sum_slices(P1sP, i, BATCH * HID);
    float c = c0[i], s = s0[i];
    float g = s * (a0[i] + p1c + b0[col]) - c * p1s + k0[col] * (2.f * s * c);
    phi0[i] -= EPSS * g;
}

__global__ void update1_kernel(float* __restrict__ phi1,
                               const float* __restrict__ c1, const float* __restrict__ s1,
                               const float* __restrict__ c2, const float* __restrict__ s2,
                               const float* __restrict__ Q1cP, const float* __restrict__ Q1sP,
                               const float* __restrict__ W2,
                               const float* __restrict__ b1, const float* __restrict__ k1)
{
    int i = blockIdx.x * blockDim.x + threadIdx.x;
    if (i >= BATCH * HID) return;
    int b = i >> 11, col = i & (HID - 1);
    // P2 = c2 @ W2, s2 @ W2  (K = 10, inlined)
    float p2c = 0.f, p2s = 0.f;
#pragma unroll
    for (int j = 0; j < OUTD; ++j) {
        float w = W2[(size_t)j * HID + col];
        p2c += w * c2[b * 16 + j];
        p2s += w * s2[b * 16 + j];
    }
    float q1c = sum_slices(Q1cP, i, BATCH * HID);
    float q1s = sum_slices(Q1sP, i, BATCH * HID);
    float c = c1[i], s = s1[i];
    float g = s * (q1c + p2c + b1[col]) - c * (q1s + p2s) + k1[col] * (2.f * s * c);
    phi1[i] -= EPSS * g;
}

// one wave per (b, j) output; fused c1@W2^T and s1@W2^T reductions
__global__ __launch_bounds__(256) void update2_kernel(
    float* __restrict__ phi2,
    const float* __restrict__ c1, const float* __restrict__ s1,
    const float* __restrict__ c2, const float* __restrict__ s2,
    const float* __restrict__ W2,
    const float* __restrict__ b2, const float* __restrict__ k2,
    const int* __restrict__ y)
{
    int wid  = (blockIdx.x * blockDim.x + threadIdx.x) >> 5;
    int lane = threadIdx.x & 31;
    if (wid >= BATCH * OUTD) return;
    int b = wid / OUTD, j = wid % OUTD;

    const float* wrow = W2 + (size_t)j * HID;
    const float* crow = c1 + (size_t)b * HID;
    const float* srow = s1 + (size_t)b * HID;
    float ac = 0.f, as = 0.f;
    for (int k = lane; k < HID; k += 32) {
        float w = wrow[k];
        ac += w * crow[k];
        as += w * srow[k];
    }
#pragma unroll
    for (int off = 16; off > 0; off >>= 1) {
        ac += __shfl_xor(ac, off, 32);
        as += __shfl_xor(as, off, 32);
    }
    if (lane == 0) {
        float cv = c2[b * 16 + j], sv = s2[b * 16 + j];
        float y1 = (y[b] == j) ? 1.f : -1.f;
        float g = sv * (ac + b2[j]) - cv * as + k2[j] * (2.f * sv * cv)
                  - BETA * (cv - y1) * sv;
        phi2[b * OUTD + j] -= EPSS * g;
    }
}

// ---------------------------------------------------------------------------
// Final concat [phi0 | phi1 | phi2] -> [256, 4106]
// ---------------------------------------------------------------------------
__global__ void concat_kernel(const float* __restrict__ phi0,
                              const float* __restrict__ phi1,
                              const float* __restrict__ phi2,
                              float* __restrict__ out)
{
    const int W = 2 * HID + OUTD;  // 4106
    int i = blockIdx.x * blockDim.x + threadIdx.x;
    if (i >= BATCH * W) return;
    int b = i / W, col = i % W;
    float v;
    if (col < HID)            v = phi0[(size_t)b * HID + col];
    else if (col < 2 * HID)   v = phi1[(size_t)b * HID + col - HID];
    else                      v = phi2[b * OUTD + col - 2 * HID];
    out[i] = v;
}

// ---------------------------------------------------------------------------
extern "C" void kernel_launch(void* const* d_in, const int* in_sizes, int n_in,
                              void* d_out, int out_size, void* d_ws, size_t ws_size,
                              hipStream_t stream)
{
    const float* x   = (const float*)d_in[0];
    const float* W0  = (const float*)d_in[1];
    const float* W1  = (const float*)d_in[2];
    const float* W2  = (const float*)d_in[3];
    const float* b0  = (const float*)d_in[4];
    const float* b1  = (const float*)d_in[5];
    const float* b2  = (const float*)d_in[6];
    const float* k0  = (const float*)d_in[7];
    const float* k1  = (const float*)d_in[8];
    const float* k2  = (const float*)d_in[9];
    const float* phi0_in = (const float*)d_in[10];
    const float* phi1_in = (const float*)d_in[11];
    const float* phi2_in = (const float*)d_in[12];
    const int*   y   = (const int*)d_in[13];
    // T = 5 (python scalar, constant)

    const size_t HB = (size_t)BATCH * HID;   // 524288
    float* w = (float*)d_ws;
    float* phi0w = w; w += HB;
    float* phi1w = w; w += HB;
    float* a0    = w; w += HB;
    float* c0    = w; w += HB;
    float* s0    = w; w += HB;
    float* c1    = w; w += HB;
    float* s1    = w; w += HB;
    float* P1cP  = w; w += SLICES * HB;      // also reused for a0 partials
    float* P1sP  = w; w += SLICES * HB;
    float* Q1cP  = w; w += SLICES * HB;
    float* Q1sP  = w; w += SLICES * HB;
    float* c2    = w; w += BATCH * 16;
    float* s2    = w; w += BATCH * 16;
    float* phi2w = w; w += BATCH * OUTD;

    // working copies of phases (inputs must stay untouched)
    hipMemcpyAsync(phi0w, phi0_in, HB * sizeof(float), hipMemcpyDeviceToDevice, stream);
    hipMemcpyAsync(phi1w, phi1_in, HB * sizeof(float), hipMemcpyDeviceToDevice, stream);
    hipMemcpyAsync(phi2w, phi2_in, (size_t)BATCH * OUTD * sizeof(float), hipMemcpyDeviceToDevice, stream);

    const dim3 gemmGrid(HID / 512, BATCH / 16, SLICES);   // (4, 16, 4)
    const dim3 gemmBlk(256);

    const int nBig  = BATCH * HID;
    const int gBig  = (nBig + 255) / 256;
    const int gC2   = (BATCH * 16 + 255) / 256;
    const int gUp2  = (BATCH * OUTD * 32 + 255) / 256;

    // a0 = x @ W0^T  (constant over steps): partials into P1cP, reduce to a0
    gemm_wmma_f32<true, false><<<gemmGrid, gemmBlk, 0, stream>>>(
        x, nullptr, W0, P1cP, nullptr, BATCH, HID, HID);
    reduce_slices_kernel<<<gBig, 256, 0, stream>>>(P1cP, a0, nBig);

    for (int t = 0; t < 5; ++t) {
        trig_kernel<<<gBig, 256, 0, stream>>>(phi0w, c0, s0, nBig);
        trig_kernel<<<gBig, 256, 0, stream>>>(phi1w, c1, s1, nBig);
        trig2_kernel<<<gC2, 256, 0, stream>>>(phi2w, c2, s2);

        // P1 = {c1,s1} @ W1      (B non-transposed, K x N row-major)
        gemm_wmma_f32<false, true><<<gemmGrid, gemmBlk, 0, stream>>>(
            c1, s1, W1, P1cP, P1sP, BATCH, HID, HID);
        // Q1 = {c0,s0} @ W1^T    (B transposed, N x K row-major)
        gemm_wmma_f32<true, true><<<gemmGrid, gemmBlk, 0, stream>>>(
            c0, s0, W1, Q1cP, Q1sP, BATCH, HID, HID);

        update0_kernel<<<gBig, 256, 0, stream>>>(phi0w, c0, s0, a0, P1cP, P1sP, b0, k0);
        update1_kernel<<<gBig, 256, 0, stream>>>(phi1w, c1, s1, c2, s2, Q1cP, Q1sP, W2, b1, k1);
        update2_kernel<<<gUp2, 256, 0, stream>>>(phi2w, c1, s1, c2, s2, W2, b2, k2, y);
    }

    const int nOut = BATCH * (2 * HID + OUTD);
    concat_kernel<<<(nOut + 255) / 256, 256, 0, stream>>>(phi0w, phi1w, phi2w, (float*)d_out);
}